// FastGuidedFilterColor_90331752169651
// MI455X (gfx1250) — compile-verified
//
#include <hip/hip_runtime.h>

// Fast guided filter (color guide), B=8, low-res 256x256, high-res 1024x1024, r=1.
// Stage 1: per low-res pixel -> A(3),b per output channel (12 planes / batch) into d_ws.
// Stage 2: bilinear x4 upsample of A,b from LDS-staged window + apply to hr_x.
// CDNA5 paths: global_load_async_to_lds_b32 (ASYNCcnt) + s_wait_asynccnt; NT hints
// on the 200 MB hr_x/out streams to preserve L2 residency of the A/b maps.

#define LH 256
#define LW 256
#define HH 1024
#define HW 1024
#define NB 8
#define LPLANE (LH * LW)
#define HPLANE ((size_t)HH * HW)
#define EPSF 1e-8f

// native clang vector (HIP's float4 is a struct and is rejected by the
// nontemporal builtins)
typedef float v4f __attribute__((ext_vector_type(4)));

// ---------- CDNA5 async global->LDS copy (tracked by ASYNCcnt) ----------
__device__ __forceinline__ void async_g2l_b32(const float* gsrc, void* lds_dst) {
  // low 32 bits of a generic LDS pointer == wave-relative LDS byte offset
  unsigned loff = (unsigned)(unsigned long long)lds_dst;
  asm volatile("global_load_async_to_lds_b32 %0, %1, off"
               :: "v"(loff), "v"(gsrc)
               : "memory");
}
__device__ __forceinline__ void async_wait_all() {
  asm volatile("s_wait_asynccnt 0" ::: "memory");
}

// =======================================================================
// Stage 1: low-res statistics -> A0,A1,A2,b for each of the 3 y-channels.
// Block = 16x16 pixels of one batch. LDS window 18x18 x 6 channels
// (3 guide + 3 target), zero-filled outside the image (truncated box sums).
// =======================================================================
__global__ __launch_bounds__(256)
void gf_stage1(const float* __restrict__ lr_x, const float* __restrict__ lr_y,
               float* __restrict__ ab /* [NB][12][LH][LW] */) {
  __shared__ float win[6 * 18 * 18];

  const int b   = blockIdx.z;
  const int gx0 = blockIdx.x * 16 - 1;
  const int gy0 = blockIdx.y * 16 - 1;
  const int tx  = threadIdx.x, ty = threadIdx.y;
  const int tid = ty * 16 + tx;

  for (int idx = tid; idx < 6 * 18 * 18; idx += 256) {
    const int plane = idx / 324;
    const int rem   = idx - plane * 324;
    const int rw    = rem / 18;
    const int cw    = rem - rw * 18;
    const int row   = gy0 + rw;
    const int col   = gx0 + cw;
    if ((unsigned)row < (unsigned)LH && (unsigned)col < (unsigned)LW) {
      const float* src = (plane < 3)
        ? lr_x + ((size_t)(b * 3 + plane)     * LH + row) * LW + col
        : lr_y + ((size_t)(b * 3 + plane - 3) * LH + row) * LW + col;
      async_g2l_b32(src, &win[idx]);
    } else {
      win[idx] = 0.0f;   // truncated box filter: outside contributes 0
    }
  }
  async_wait_all();
  __syncthreads();

  const int x = blockIdx.x * 16 + tx;
  const int y = blockIdx.y * 16 + ty;

  float sx[3]  = {0.f, 0.f, 0.f};
  float sy[3]  = {0.f, 0.f, 0.f};
  float sxx[6] = {0.f, 0.f, 0.f, 0.f, 0.f, 0.f};
  float sxy[9] = {0.f, 0.f, 0.f, 0.f, 0.f, 0.f, 0.f, 0.f, 0.f};

#pragma unroll
  for (int dy = 0; dy < 3; ++dy) {
#pragma unroll
    for (int dx = 0; dx < 3; ++dx) {
      const int r = ty + dy, c = tx + dx;
      float xv[3], yv[3];
#pragma unroll
      for (int ch = 0; ch < 3; ++ch) {
        xv[ch] = win[(ch * 18 + r) * 18 + c];
        yv[ch] = win[((ch + 3) * 18 + r) * 18 + c];
        sx[ch] += xv[ch];
        sy[ch] += yv[ch];
      }
      sxx[0] += xv[0] * xv[0];
      sxx[1] += xv[0] * xv[1];
      sxx[2] += xv[0] * xv[2];
      sxx[3] += xv[1] * xv[1];
      sxx[4] += xv[1] * xv[2];
      sxx[5] += xv[2] * xv[2];
#pragma unroll
      for (int cc = 0; cc < 3; ++cc)
#pragma unroll
        for (int ii = 0; ii < 3; ++ii)
          sxy[cc * 3 + ii] += xv[cc] * yv[ii];
    }
  }

  // window element count (truncated at borders)
  const int ymin = (y > 0) ? y - 1 : 0, ymax = (y < LH - 1) ? y + 1 : LH - 1;
  const int xmin = (x > 0) ? x - 1 : 0, xmax = (x < LW - 1) ? x + 1 : LW - 1;
  const float fN = (float)((ymax - ymin + 1) * (xmax - xmin + 1));

  float mx[3], my[3];
#pragma unroll
  for (int ch = 0; ch < 3; ++ch) { mx[ch] = sx[ch] / fN; my[ch] = sy[ch] / fN; }

  const float v00 = sxx[0] / fN - mx[0] * mx[0] + EPSF;
  const float v01 = sxx[1] / fN - mx[0] * mx[1] + EPSF;
  const float v02 = sxx[2] / fN - mx[0] * mx[2] + EPSF;
  const float v11 = sxx[3] / fN - mx[1] * mx[1] + EPSF;
  const float v12 = sxx[4] / fN - mx[1] * mx[2] + EPSF;
  const float v22 = sxx[5] / fN - mx[2] * mx[2] + EPSF;

  // cofactors of the symmetric 3x3 covariance
  float i0 = v11 * v22 - v12 * v12;
  float i1 = v12 * v02 - v01 * v22;
  float i2 = v01 * v12 - v11 * v02;
  float i3 = v00 * v22 - v02 * v02;
  float i4 = v02 * v01 - v00 * v12;
  float i5 = v00 * v11 - v01 * v01;
  const float det = i0 * v00 + i1 * v01 + i2 * v02;
  i0 /= det; i1 /= det; i2 /= det; i3 /= det; i4 /= det; i5 /= det;

#pragma unroll
  for (int i = 0; i < 3; ++i) {
    const float c0 = sxy[0 * 3 + i] / fN - mx[0] * my[i];
    const float c1 = sxy[1 * 3 + i] / fN - mx[1] * my[i];
    const float c2 = sxy[2 * 3 + i] / fN - mx[2] * my[i];
    const float A0 = i0 * c0 + i1 * c1 + i2 * c2;
    const float A1 = i1 * c0 + i3 * c1 + i4 * c2;
    const float A2 = i2 * c0 + i4 * c1 + i5 * c2;
    const float bb = my[i] - (A0 * mx[0] + A1 * mx[1] + A2 * mx[2]);
    const size_t o = (((size_t)b * 12 + i * 4) * LH + y) * LW + x;
    ab[o]               = A0;
    ab[o + 1 * LPLANE]  = A1;
    ab[o + 2 * LPLANE]  = A2;
    ab[o + 3 * LPLANE]  = bb;
  }
}

// =======================================================================
// Stage 2: bilinear x4 upsample of A,b + apply to hr_x.
// Block = 256 threads covering a 128x8 output tile of one batch; each
// thread produces 4 consecutive X pixels (v4f I/O) for all 3 channels.
// Low-res window: 12 planes x 4 rows x 34 cols, async-staged with
// clamp-to-edge indices (== JAX resize boundary renormalization).
// Vertical blend first (matches JAX's per-dimension resize order), then
// the 4 fixed horizontal phases: 14 VALU ops/plane instead of 24.
// =======================================================================
__device__ __forceinline__ v4f bilerp4(const float* __restrict__ w,
                                       int plane, int wr0, int wc,
                                       float fy, float fy1) {
  const float* r0 = w + (plane * 4 + wr0) * 34 + (wc - 1);
  const float* r1 = r0 + 34;
  const float v0 = fy1 * r0[0] + fy * r1[0];
  const float v1 = fy1 * r0[1] + fy * r1[1];
  const float v2 = fy1 * r0[2] + fy * r1[2];
  // horizontal phases: fx = 0.625, 0.875 (pair q-1,q); 0.125, 0.375 (pair q,q+1)
  v4f r = {0.375f * v0 + 0.625f * v1,
           0.125f * v0 + 0.875f * v1,
           0.875f * v1 + 0.125f * v2,
           0.625f * v1 + 0.375f * v2};
  return r;
}

__global__ __launch_bounds__(256)
void gf_stage2(const float* __restrict__ ab, const float* __restrict__ hr_x,
               float* __restrict__ out) {
  __shared__ float win[12 * 4 * 34];

  const int b   = blockIdx.z;
  const int lx0 = blockIdx.x * 32 - 1;  // low-res col of window col 0
  const int ly0 = blockIdx.y * 2  - 1;  // low-res row of window row 0
  const int tid = threadIdx.x;

  for (int idx = tid; idx < 12 * 4 * 34; idx += 256) {
    const int plane = idx / 136;
    const int rem   = idx - plane * 136;
    const int rw    = rem / 34;
    const int cw    = rem - rw * 34;
    int row = ly0 + rw; row = (row < 0) ? 0 : ((row > LH - 1) ? LH - 1 : row);
    int col = lx0 + cw; col = (col < 0) ? 0 : ((col > LW - 1) ? LW - 1 : col);
    const float* src = ab + (((size_t)b * 12 + plane) * LH + row) * LW + col;
    async_g2l_b32(src, &win[idx]);
  }
  async_wait_all();
  __syncthreads();

  const int qx   = tid & 31;            // 4-pixel group within tile (lane id)
  const int rowt = tid >> 5;            // 0..7, uniform per wave32
  const int X = blockIdx.x * 128 + qx * 4;
  const int Y = blockIdx.y * 8 + rowt;
  const int s = rowt & 3;               // vertical phase (wave-uniform)
  const int wr0 = (rowt >> 2) + ((s >> 1) & 1);           // window row of y0
  const float fy  = 0.125f + 0.25f * (float)((s + 2) & 3); // 0.625,0.875,0.125,0.375
  const float fy1 = 1.0f - fy;
  const int wc = qx + 1;                // window col of q (lane-consecutive)

  const size_t hrbase = ((size_t)(b * 3) * HH + Y) * HW + X;
  const v4f h0 = __builtin_nontemporal_load((const v4f*)(hr_x + hrbase));
  const v4f h1 = __builtin_nontemporal_load((const v4f*)(hr_x + hrbase + HPLANE));
  const v4f h2 = __builtin_nontemporal_load((const v4f*)(hr_x + hrbase + 2 * HPLANE));

#pragma unroll
  for (int i = 0; i < 3; ++i) {
    const v4f A0 = bilerp4(win, i * 4 + 0, wr0, wc, fy, fy1);
    const v4f A1 = bilerp4(win, i * 4 + 1, wr0, wc, fy, fy1);
    const v4f A2 = bilerp4(win, i * 4 + 2, wr0, wc, fy, fy1);
    const v4f Bb = bilerp4(win, i * 4 + 3, wr0, wc, fy, fy1);
    v4f o = A0 * h0 + A1 * h1 + A2 * h2 + Bb;
    __builtin_nontemporal_store(o, (v4f*)(out + (((size_t)(b * 3 + i)) * HH + Y) * HW + X));
  }
}

extern "C" void kernel_launch(void* const* d_in, const int* in_sizes, int n_in,
                              void* d_out, int out_size, void* d_ws, size_t ws_size,
                              hipStream_t stream) {
  const float* lr_x = (const float*)d_in[0];
  const float* lr_y = (const float*)d_in[1];
  const float* hr_x = (const float*)d_in[2];
  // d_in[3] is r == 1, baked in at compile time.
  float* ab  = (float*)d_ws;   // [NB][12][256][256] floats = 24 MB
  float* out = (float*)d_out;

  dim3 b1(16, 16, 1), g1(LW / 16, LH / 16, NB);
  gf_stage1<<<g1, b1, 0, stream>>>(lr_x, lr_y, ab);

  dim3 b2(256, 1, 1), g2(HW / 128, HH / 8, NB);
  gf_stage2<<<g2, b2, 0, stream>>>(ab, hr_x, out);
}